// MutualInformation_48954037239853
// MI455X (gfx1250) — compile-verified
//
#include <hip/hip_runtime.h>
#include <hip/hip_bf16.h>

// ---------------------------------------------------------------------------
// MutualInformation for MI455X (gfx1250).
// Pure bandwidth problem: 128 MiB of fp32 reads -> ~5.6us floor @ 23.3 TB/s.
// Strategy: async global->LDS staging (ASYNCcnt) + wave32 wave-private LDS
// histograms (ds_add_u32, no barriers on the hot path) + tiny MI epilogue.
// hist_kernel is defined FIRST so the disasm snippet shows the async copies.
// ---------------------------------------------------------------------------

#define NB      64                  // histogram bins
#define NBATCH  64                  // batches
#define NPB     (512 * 512)         // elements per batch
#define BPB     64                  // blocks per batch
#define CHUNK   (NPB / BPB)         // 4096 elements per block
#define NT      256                 // threads per block (8 waves on wave32)
#define NWAVE   (NT / 32)           // 8
#define EPT     (CHUNK / NT)        // 16 elements per thread
#define EPW     (CHUNK / NWAVE)     // 512 elements per wave

// Builtin signature (from clang diagnostic): param0 is addrspace(1) int4*,
// param1 is addrspace(3) int4*, then imm offset, imm cpol.
typedef int v4i __attribute__((ext_vector_type(4)));
typedef __attribute__((address_space(1))) v4i gv4i;   // global int4
typedef __attribute__((address_space(3))) v4i lv4i;   // LDS int4

// One wave-wide async copy: 32 lanes x 16B = 512B global -> LDS. Tracked by
// ASYNCcnt, returns no data to VGPRs.
__device__ __forceinline__ void async_copy_b128(const float* g, float* l) {
#if __has_builtin(__builtin_amdgcn_global_load_async_to_lds_b128)
    __builtin_amdgcn_global_load_async_to_lds_b128((gv4i*)g, (lv4i*)l,
                                                   /*offset=*/0, /*cpol=*/0);
#else
    unsigned lds_off = (unsigned)(size_t)(lv4i*)l;
    unsigned long long ga = (unsigned long long)g;
    asm volatile("global_load_async_to_lds_b128 %0, %1, off"
                 :: "v"(lds_off), "v"(ga) : "memory");
#endif
}

__device__ __forceinline__ void wait_async0() {
#if __has_builtin(__builtin_amdgcn_s_wait_asynccnt)
    __builtin_amdgcn_s_wait_asynccnt(0);
#else
    asm volatile("s_wait_asynccnt 0" ::: "memory");
#endif
    asm volatile("" ::: "memory");  // keep compiler from hoisting LDS reads
}

__device__ __forceinline__ void bin_add(unsigned* whist, float xv, float yv) {
    // v = x*64 + y; histc(bins=64, min=0, max=64): floor binning, v==64 -> last
    // bin, values outside [0,64] dropped.
    float v = xv * 64.0f + yv;
    if (v >= 0.0f && v <= 64.0f) {
        int i = (int)v;            // v >= 0, so truncation == floor
        if (i > NB - 1) i = NB - 1;
        atomicAdd(&whist[i], 1u);  // ds_add_u32 (wave-private copy)
    }
}

__global__ void __launch_bounds__(NT)
hist_kernel(const float* __restrict__ x, const float* __restrict__ y,
            unsigned* __restrict__ ghist) {
    __shared__ __align__(16) float sx[CHUNK];      // 16 KB
    __shared__ __align__(16) float sy[CHUNK];      // 16 KB
    __shared__ unsigned shist[NWAVE * NB];         //  2 KB

    const int b    = blockIdx.y;
    const int blk  = blockIdx.x;
    const int tid  = threadIdx.x;
    const int wave = tid >> 5;
    const int lane = tid & 31;

    // Zero this wave's private histogram (only this wave touches it -> no
    // cross-wave barrier needed before accumulation).
    shist[wave * NB + lane]      = 0u;
    shist[wave * NB + 32 + lane] = 0u;

    const size_t gbase = (size_t)b * NPB + (size_t)blk * CHUNK
                       + (size_t)wave * EPW;
    const int    lbase = wave * EPW;

    // Stage this wave's 512 x + 512 y elements: 8 outstanding async B128
    // copies per wave before a single s_wait_asynccnt 0.
#pragma unroll
    for (int i = 0; i < EPW / 128; ++i) {          // 4 iterations
        const int o = i * 128 + lane * 4;          // 16B per lane per copy
        async_copy_b128(x + gbase + o, sx + lbase + o);
        async_copy_b128(y + gbase + o, sy + lbase + o);
    }
    wait_async0();

    // Consume from LDS (wave reads only what it staged itself).
    const float4* sx4 = (const float4*)(sx + lbase);
    const float4* sy4 = (const float4*)(sy + lbase);
    unsigned* whist = shist + wave * NB;
#pragma unroll
    for (int i = 0; i < EPT / 4; ++i) {            // 4 float4 per lane
        float4 xv = sx4[i * 32 + lane];
        float4 yv = sy4[i * 32 + lane];
        bin_add(whist, xv.x, yv.x);
        bin_add(whist, xv.y, yv.y);
        bin_add(whist, xv.z, yv.z);
        bin_add(whist, xv.w, yv.w);
    }

    __syncthreads();  // only barrier: combine the 8 wave-private histograms
    if (tid < NB) {
        unsigned s = 0;
#pragma unroll
        for (int w = 0; w < NWAVE; ++w) s += shist[w * NB + tid];
        atomicAdd(&ghist[b * NB + tid], s);        // global_atomic_add_u32
    }
}

__global__ void zero_kernel(unsigned* __restrict__ p, int n) {
    int i = blockIdx.x * blockDim.x + threadIdx.x;
    if (i < n) p[i] = 0u;
}

__global__ void __launch_bounds__(NBATCH)
mi_kernel(const unsigned* __restrict__ ghist, float* __restrict__ out) {
    __shared__ float smi[NBATCH];
    const int b = threadIdx.x;                      // one batch per lane
    const unsigned* h = ghist + b * NB;

    float total = 0.0f;
    for (int i = 0; i < NB; ++i) total += (float)h[i];
    const float inv = 1.0f / total;

    // px = sum_i (h_i/total + 1e-8), computed by summation like the reference
    float px = 0.0f;
    for (int i = 0; i < NB; ++i) px += (float)h[i] * inv + 1e-8f;
    const float d = px * px;

    float mi = 0.0f;
    for (int i = 0; i < NB; ++i) {
        float p = (float)h[i] * inv + 1e-8f;
        mi += p * logf(p / d);
    }
    smi[b] = mi;
    __syncthreads();
    if (b == 0) {
        float acc = 0.0f;
        for (int i = 0; i < NBATCH; ++i) acc += smi[i];
        out[0] = -acc / (float)NBATCH;
    }
}

extern "C" void kernel_launch(void* const* d_in, const int* in_sizes, int n_in,
                              void* d_out, int out_size, void* d_ws, size_t ws_size,
                              hipStream_t stream) {
    const float* x = (const float*)d_in[0];
    const float* y = (const float*)d_in[1];
    unsigned* ghist = (unsigned*)d_ws;              // 64*64 u32 = 16 KB scratch
    float* out = (float*)d_out;

    zero_kernel<<<(NBATCH * NB + 255) / 256, 256, 0, stream>>>(ghist, NBATCH * NB);
    hist_kernel<<<dim3(BPB, NBATCH), NT, 0, stream>>>(x, y, ghist);
    mi_kernel<<<1, NBATCH, 0, stream>>>(ghist, out);
}